// HGT_73237782331619
// MI455X (gfx1250) — compile-verified
//
#include <hip/hip_runtime.h>
#include <math.h>

typedef float v2f __attribute__((ext_vector_type(2)));
typedef float v8f __attribute__((ext_vector_type(8)));

// ---------------- device helpers ----------------

__device__ inline v8f wmma4f32(v2f a, v2f b, v8f c) {
  // D = A(16x4) * B(4x16) + C(16x16), fp32, wave32
  return __builtin_amdgcn_wmma_f32_16x16x4_f32(false, a, false, b, (short)0, c,
                                               false, false);
}

__device__ inline float geluf(float x) {
  return 0.5f * x * (1.0f + erff(x * 0.70710678118654752f));
}

__device__ inline void atomicMaxF(float* addr, float val) {
  int* ai = reinterpret_cast<int*>(addr);
  int cur = *ai;
  while (__int_as_float(cur) < val) {
    int prev = atomicCAS(ai, cur, __float_as_int(val));
    if (prev == cur) break;
    cur = prev;
  }
}

__device__ inline void atomicAddF(float* addr, float v) {
  unsafeAtomicAdd(addr, v);  // lowers to global_atomic_add_f32
}

// ---------------- generic WMMA GEMM (W staged through LDS) ----------------
// Y[r, col] = act( X[r,:Din] @ W[Din,Dout] + bias[col] )
// actMode: 0=none, 1=relu, 2 = gelu applied to X elements, then
//          Y = alpha*(..) + (1-alpha)*skipX  with alpha = sigmoid(*skipParam)
// One wave computes a 16-row x 64-col tile. Dout must be a multiple of 64.
// Din*Dout <= 12288 (largest here: 64x192 for the fused KQV projection).

__global__ __launch_bounds__(256) void k_gemm_wmma(
    const float* __restrict__ X, int ldx, const float* __restrict__ W,
    const float* __restrict__ bias, float* __restrict__ Y, int ldy,
    const float* __restrict__ skipX, const float* __restrict__ skipParam,
    int N, int Din, int Dout, int actMode) {
  __shared__ float sW[12288];
  const int nW = Din * Dout;
  for (int i = threadIdx.x; i < nW; i += 256) sW[i] = W[i];
  __syncthreads();

  const int chunks = Dout >> 6;
  const int nTiles = (N + 15) >> 4;
  const int wid = blockIdx.x * 8 + (threadIdx.x >> 5);
  if (wid >= nTiles * chunks) return;  // wave-uniform, after the only barrier
  const int rowTile = wid / chunks;
  const int chunk = wid % chunks;
  const int lane = threadIdx.x & 31;
  const int half = lane >> 4;
  const int l16 = lane & 15;

  const int arow = rowTile * 16 + l16;  // A-fragment row for this lane
  const bool rOK = (arow < N);
  const float* xr = X + (size_t)arow * ldx;
  const int colBase = chunk * 64 + l16;

  v8f acc[4] = {};
#pragma unroll 4
  for (int k0 = 0; k0 < Din; k0 += 4) {
    const int ka = k0 + 2 * half;  // this lane's K pair: ka, ka+1
    v2f a;
    if (rOK) {
      float ax = xr[ka], ay = xr[ka + 1];
      if (actMode == 2) {
        ax = geluf(ax);
        ay = geluf(ay);
      }
      a.x = ax;
      a.y = ay;
    } else {
      a.x = 0.0f;
      a.y = 0.0f;
    }
    const float* w0 = sW + ka * Dout + colBase;
    const float* w1 = w0 + Dout;
    v2f b;
    b.x = w0[0];
    b.y = w1[0];
    acc[0] = wmma4f32(a, b, acc[0]);
    b.x = w0[16];
    b.y = w1[16];
    acc[1] = wmma4f32(a, b, acc[1]);
    b.x = w0[32];
    b.y = w1[32];
    acc[2] = wmma4f32(a, b, acc[2]);
    b.x = w0[48];
    b.y = w1[48];
    acc[3] = wmma4f32(a, b, acc[3]);
  }

  float alpha = 1.0f, beta = 0.0f;
  if (actMode == 2) {
    const float sp = *skipParam;
    alpha = 1.0f / (1.0f + expf(-sp));
    beta = 1.0f - alpha;
  }
#pragma unroll
  for (int c = 0; c < 4; ++c) {
    const int col = chunk * 64 + c * 16 + l16;
    const float bv = bias[col];
#pragma unroll
    for (int r8 = 0; r8 < 8; ++r8) {
      const int r = rowTile * 16 + r8 + 8 * half;  // C layout: M = r8 + 8*half
      if (r < N) {
        float v = acc[c][r8] + bv;
        if (actMode == 1) v = v > 0.0f ? v : 0.0f;
        if (actMode == 2) v = alpha * v + beta * skipX[(size_t)r * 64 + col];
        Y[(size_t)r * ldy + col] = v;
      }
    }
  }
}

// ---------------- per-head 16x16 relation transform ----------------
// Out[n, h*16+e] = sum_d In[n*ldin + colOff + h*16 + d] * Rel[h*256 + d*16 + e]
__global__ __launch_bounds__(256) void k_rel_wmma(
    const float* __restrict__ In, int ldin, int colOff,
    const float* __restrict__ Rel, float* __restrict__ Out, int N) {
  __shared__ float sRel[1024];  // 4 heads x 16x16
  for (int i = threadIdx.x; i < 1024; i += 256) sRel[i] = Rel[i];
  __syncthreads();

  const int nTiles = (N + 15) >> 4;
  const int wid = blockIdx.x * 8 + (threadIdx.x >> 5);
  if (wid >= nTiles * 4) return;
  const int rowTile = wid >> 2;
  const int h = wid & 3;
  const int lane = threadIdx.x & 31;
  const int half = lane >> 4;
  const int l16 = lane & 15;

  const int arow = rowTile * 16 + l16;
  const bool rOK = (arow < N);
  const float* xr = In + (size_t)arow * ldin + colOff + h * 16;
  const float* rel = sRel + h * 256;

  v8f acc = {};
#pragma unroll
  for (int k0 = 0; k0 < 16; k0 += 4) {
    const int ka = k0 + 2 * half;
    v2f a;
    if (rOK) {
      a.x = xr[ka];
      a.y = xr[ka + 1];
    } else {
      a.x = 0.0f;
      a.y = 0.0f;
    }
    v2f b;
    b.x = rel[ka * 16 + l16];
    b.y = rel[(ka + 1) * 16 + l16];
    acc = wmma4f32(a, b, acc);
  }
#pragma unroll
  for (int r8 = 0; r8 < 8; ++r8) {
    const int r = rowTile * 16 + r8 + 8 * half;
    if (r < N) Out[(size_t)r * 64 + h * 16 + l16] = acc[r8];
  }
}

// ---------------- init (softmax state + aggregation buffer) ----------------
__global__ void k_init(float* __restrict__ maxb, float* __restrict__ sumb,
                       float* __restrict__ agg, int n4, int n64) {
  const int i = blockIdx.x * 256 + threadIdx.x;
  if (i < n4) {
    maxb[i] = -INFINITY;
    sumb[i] = 0.0f;
  }
  if (i < n64) agg[i] = 0.0f;
}

// ---------------- edge logits + running max (one wave per edge) ----------
// lane covers elements {lane, lane+32} of the 64-float rows:
//   lane in [0,16):  head0 (p0) and head2 (p1)
//   lane in [16,32): head1 (p0) and head3 (p1)
__global__ __launch_bounds__(256) void k_logits(
    const float* __restrict__ ke, const float* __restrict__ kqvDst,
    const int* __restrict__ ei, int E, const float* __restrict__ pRel,
    float* __restrict__ logit, float* __restrict__ maxb) {
  const int e = blockIdx.x * 8 + (threadIdx.x >> 5);
  if (e >= E) return;
  const int lane = threadIdx.x & 31;
  const int s = ei[e];
  const int d = ei[E + e];
  const float* kr = ke + (size_t)s * 64;
  const float* qr = kqvDst + (size_t)d * 192 + 64;  // q slice (256B aligned)
  float p0 = kr[lane] * qr[lane];
  float p1 = kr[lane + 32] * qr[lane + 32];
#pragma unroll
  for (int off = 8; off >= 1; off >>= 1) {  // reduce within 16-lane groups
    p0 += __shfl_xor(p0, off, 32);
    p1 += __shfl_xor(p1, off, 32);
  }
  if ((lane & 15) == 0) {
    const int hb = lane >> 4;  // 0 or 1
    const float a0 = p0 * pRel[hb] * 0.25f;      // 1/sqrt(16)
    const float a1 = p1 * pRel[hb + 2] * 0.25f;
    logit[(size_t)e * 4 + hb] = a0;
    logit[(size_t)e * 4 + hb + 2] = a1;
    atomicMaxF(&maxb[(size_t)d * 4 + hb], a0);
    atomicMaxF(&maxb[(size_t)d * 4 + hb + 2], a1);
  }
}

// ---------------- exp(a - max) + running sum ----------------
__global__ void k_expsum(const int* __restrict__ ei, int E,
                         float* __restrict__ logit,
                         const float* __restrict__ maxb,
                         float* __restrict__ sumb) {
  const int idx = blockIdx.x * 256 + threadIdx.x;
  const int e = idx >> 2;
  const int h = idx & 3;
  if (e >= E) return;
  const int d = ei[E + e];
  const float v = expf(logit[(size_t)e * 4 + h] - maxb[(size_t)d * 4 + h]);
  logit[(size_t)e * 4 + h] = v;
  atomicAddF(&sumb[(size_t)d * 4 + h], v);
}

// ---------------- weighted message scatter-add (one wave per edge) -------
__global__ __launch_bounds__(256) void k_agg(
    const int* __restrict__ ei, int E, const float* __restrict__ logit,
    const float* __restrict__ sumb, const float* __restrict__ ve,
    float* __restrict__ agg) {
  const int e = blockIdx.x * 8 + (threadIdx.x >> 5);
  if (e >= E) return;
  const int lane = threadIdx.x & 31;
  const int s = ei[e];
  const int d = ei[E + e];
  const float* vr = ve + (size_t)s * 64;
  float* ar = agg + (size_t)d * 64;
  const int h0 = lane >> 4;  // head of element `lane`; element lane+32 -> h0+2
  const float w0 = logit[(size_t)e * 4 + h0] / sumb[(size_t)d * 4 + h0];
  const float w1 =
      logit[(size_t)e * 4 + h0 + 2] / sumb[(size_t)d * 4 + h0 + 2];
  atomicAddF(&ar[lane], vr[lane] * w0);
  atomicAddF(&ar[lane + 32], vr[lane + 32] * w1);
}

// ---------------- host orchestration ----------------

extern "C" void kernel_launch(void* const* d_in, const int* in_sizes, int n_in,
                              void* d_out, int out_size, void* d_ws,
                              size_t ws_size, hipStream_t stream) {
  (void)in_sizes;
  (void)n_in;
  (void)out_size;
  (void)ws_size;

  // storage order == output order: author, paper, term, conference
  const int NN[4] = {50000, 100000, 10000, 50};
  const int NOFF[4] = {0, 50000, 150000, 160000};
  const int TOT = 160050;
  // jax tree-flatten sorts dict keys: author, conference, paper, term
  const int PSORT[4] = {0, 2, 3, 1};

  struct EDesc { int ei, E, st, dt, rel, logOff; };
  // rel = index into sorted edge-key order:
  // author__paper=0, conference__paper=1, paper__author=2,
  // paper__conference=3, paper__term=4, term__paper=5
  const EDesc ED[6] = {
      {4, 300000, 0, 1, 0, 0},        // author  -> paper
      {5, 300000, 1, 0, 2, 300000},   // paper   -> author
      {6, 500000, 1, 2, 4, 600000},   // paper   -> term
      {7, 500000, 2, 1, 5, 1100000},  // term    -> paper
      {8, 100000, 1, 3, 3, 1600000},  // paper   -> conference
      {9, 100000, 3, 1, 1, 1700000},  // conference -> paper
  };

  auto F = [&](int i) { return (const float*)d_in[i]; };
  auto I = [&](int i) { return (const int*)d_in[i]; };

  float* ws = (float*)d_ws;
  size_t o = 0;
  float* X0 = ws + o;   o += (size_t)TOT * 64;
  float* X1 = ws + o;   o += (size_t)TOT * 64;
  float* KQV = ws + o;  o += (size_t)TOT * 192;
  float* REL = ws + o;  o += (size_t)100000 * 64;
  float* LOG = ws + o;  o += (size_t)1800000 * 4;
  float* MAXB = ws + o; o += (size_t)TOT * 4;
  float* SUMB = ws + o; o += (size_t)TOT * 4;
  float* AGG = ws + o;  o += (size_t)TOT * 64;

  // param leaf indices (flattened pytree appended after the 10 arrays)
  const int P0 = 10;  // in_lin: (b,w) x {author,conference,paper,term}
  // per-layer block (stride 38): a_rel[6] kqv[8] m_rel[6] out[8] p_rel[6] skip[4]
  auto LBASE = [&](int l) { return P0 + 8 + 38 * l; };

  // ---- input projection: X0 = relu(x @ Win + b) ----
  for (int t = 0; t < 4; ++t) {
    const int N = NN[t];
    const int waves = (N + 15) / 16;  // Dout=64 -> 1 chunk
    k_gemm_wmma<<<(waves + 7) / 8, 256, 0, stream>>>(
        F(t), 128, F(P0 + 2 * PSORT[t] + 1), F(P0 + 2 * PSORT[t]),
        X0 + (size_t)NOFF[t] * 64, 64, nullptr, nullptr, N, 128, 64, 1);
  }

  for (int l = 0; l < 2; ++l) {
    const int L = LBASE(l);
    const float* xin = (l == 0) ? X0 : X1;
    float* xout = (l == 0) ? X1 : (float*)d_out;

    // fused KQV projection: KQV = xin @ Wkqv + b  (k|q|v slices)
    for (int t = 0; t < 4; ++t) {
      const int N = NN[t];
      const int waves = ((N + 15) / 16) * 3;  // Dout=192 -> 3 chunks
      k_gemm_wmma<<<(waves + 7) / 8, 256, 0, stream>>>(
          xin + (size_t)NOFF[t] * 64, 64, F(L + 6 + 2 * PSORT[t] + 1),
          F(L + 6 + 2 * PSORT[t]), KQV + (size_t)NOFF[t] * 192, 192, nullptr,
          nullptr, N, 64, 192, 0);
    }

    // reset softmax state + aggregation buffer
    {
      const int n64 = TOT * 64, n4 = TOT * 4;
      k_init<<<(n64 + 255) / 256, 256, 0, stream>>>(MAXB, SUMB, AGG, n4, n64);
    }

    // pass 1: ke = k @ a_rel, edge logits + per-dst max
    for (int e = 0; e < 6; ++e) {
      const EDesc& ed = ED[e];
      const int Ns = NN[ed.st];
      const int waves = ((Ns + 15) / 16) * 4;
      k_rel_wmma<<<(waves + 7) / 8, 256, 0, stream>>>(
          KQV + (size_t)NOFF[ed.st] * 192, 192, 0, F(L + 0 + ed.rel), REL, Ns);
      k_logits<<<(ed.E + 7) / 8, 256, 0, stream>>>(
          REL, KQV + (size_t)NOFF[ed.dt] * 192, I(ed.ei), ed.E,
          F(L + 28 + ed.rel), LOG + (size_t)ed.logOff * 4,
          MAXB + (size_t)NOFF[ed.dt] * 4);
    }

    // pass 2: exponentiate + per-dst sum
    for (int e = 0; e < 6; ++e) {
      const EDesc& ed = ED[e];
      const int n = ed.E * 4;
      k_expsum<<<(n + 255) / 256, 256, 0, stream>>>(
          I(ed.ei), ed.E, LOG + (size_t)ed.logOff * 4,
          MAXB + (size_t)NOFF[ed.dt] * 4, SUMB + (size_t)NOFF[ed.dt] * 4);
    }

    // pass 3: ve = v @ m_rel, weighted scatter-add into AGG
    for (int e = 0; e < 6; ++e) {
      const EDesc& ed = ED[e];
      const int Ns = NN[ed.st];
      const int waves = ((Ns + 15) / 16) * 4;
      k_rel_wmma<<<(waves + 7) / 8, 256, 0, stream>>>(
          KQV + (size_t)NOFF[ed.st] * 192, 192, 128, F(L + 14 + ed.rel), REL,
          Ns);
      k_agg<<<(ed.E + 7) / 8, 256, 0, stream>>>(
          I(ed.ei), ed.E, LOG + (size_t)ed.logOff * 4,
          SUMB + (size_t)NOFF[ed.dt] * 4, REL,
          AGG + (size_t)NOFF[ed.dt] * 64);
    }

    // out projection: xout = alpha*(gelu(AGG) @ Wout + b) + (1-alpha)*xin
    for (int t = 0; t < 4; ++t) {
      const int N = NN[t];
      const int waves = (N + 15) / 16;
      k_gemm_wmma<<<(waves + 7) / 8, 256, 0, stream>>>(
          AGG + (size_t)NOFF[t] * 64, 64, F(L + 20 + 2 * PSORT[t] + 1),
          F(L + 20 + 2 * PSORT[t]), xout + (size_t)NOFF[t] * 64, 64,
          xin + (size_t)NOFF[t] * 64, F(L + 34 + PSORT[t]), N, 64, 64, 2);
    }
  }
}